// iPEPS_30116310679732
// MI455X (gfx1250) — compile-verified
//
#include <hip/hip_runtime.h>
#include <hip/hip_bf16.h>
#include <math.h>

typedef __attribute__((ext_vector_type(2))) float v2f;
typedef __attribute__((ext_vector_type(8))) float v8f;

#define KC 32   // K-chunk staged in LDS per block iteration

// ---------------------------------------------------------------------------
// Core WMMA f32 GEMM:  C(MxN) = A(MxK) @ B(KxN), all row-major f32.
// One block = 64x64 C tile (4 waves x 16 rows, sharing one B panel in LDS).
// B panel staged via gfx1250 async global->LDS (ASYNCcnt), consumed by
// v_wmma_f32_16x16x4_f32 (4 per K-step).  Requires M%64==0, N%64==0, K%32==0.
// ---------------------------------------------------------------------------
__global__ __launch_bounds__(128) void k_gemm(const float* __restrict__ A,
                                              const float* __restrict__ B,
                                              float* __restrict__ C,
                                              int M, int N, int K) {
    __shared__ float Bs[KC * 64];
    const int tid  = threadIdx.x;
    const int wave = tid >> 5;
    const int lane = tid & 31;
    const int nbN  = N >> 6;
    const int bm   = blockIdx.x / nbN;
    const int bn   = blockIdx.x % nbN;
    const int n0   = bn << 6;
    const int r    = lane & 15;
    const int hi   = lane >> 4;                 // 0: K pair {0,1}, 1: K pair {2,3}

    v8f acc0 = {}, acc1 = {}, acc2 = {}, acc3 = {};
    const float* Arow = A + (size_t)((bm << 6) + (wave << 4) + r) * K + (hi ? 2 : 0);

    // per-thread slice of the async B-panel copy: 4 x b128 per chunk
    const int srow = tid >> 4;                  // 0..7, stride 8 over 32 rows
    const int sseg = (tid & 15) << 2;           // float4 segment within a 64-wide row
    const unsigned ldsbase = (unsigned)(size_t)(&Bs[0]);   // LDS addr = low 32 bits

    for (int k0 = 0; k0 < K; k0 += KC) {
        // stage B[k0..k0+31][n0..n0+63] into LDS with async loads
#pragma unroll
        for (int i = 0; i < 4; ++i) {
            int row = srow + i * 8;
            const float* g = B + (size_t)(k0 + row) * N + n0 + sseg;
            unsigned l = ldsbase + (unsigned)((row << 6) + sseg) * 4u;
            asm volatile("global_load_async_to_lds_b128 %0, %1, off"
                         :: "v"(l), "v"(g) : "memory");
        }
        asm volatile("s_wait_asynccnt 0x0" ::: "memory");
        __syncthreads();

#pragma unroll
        for (int kk = 0; kk < KC; kk += 4) {
            v2f af;
            af.x = Arow[k0 + kk];
            af.y = Arow[k0 + kk + 1];
            const float* Bk = &Bs[(kk + (hi ? 2 : 0)) << 6] + r;
            v2f b0, b1, b2, b3;
            b0.x = Bk[0];   b0.y = Bk[64];
            b1.x = Bk[16];  b1.y = Bk[80];
            b2.x = Bk[32];  b2.y = Bk[96];
            b3.x = Bk[48];  b3.y = Bk[112];
            acc0 = __builtin_amdgcn_wmma_f32_16x16x4_f32(false, af, false, b0, (short)0, acc0, false, false);
            acc1 = __builtin_amdgcn_wmma_f32_16x16x4_f32(false, af, false, b1, (short)0, acc1, false, false);
            acc2 = __builtin_amdgcn_wmma_f32_16x16x4_f32(false, af, false, b2, (short)0, acc2, false, false);
            acc3 = __builtin_amdgcn_wmma_f32_16x16x4_f32(false, af, false, b3, (short)0, acc3, false, false);
        }
        __syncthreads();
    }

    // C layout: VGPR v -> rows +v (lanes 0-15) / +8+v (lanes 16-31), col = n0+16j+r
    float* Crow = C + (size_t)((bm << 6) + (wave << 4) + (hi ? 8 : 0)) * N + n0 + r;
#pragma unroll
    for (int v = 0; v < 8; ++v) {
        Crow[(size_t)v * N + 0]  = acc0[v];
        Crow[(size_t)v * N + 16] = acc1[v];
        Crow[(size_t)v * N + 32] = acc2[v];
        Crow[(size_t)v * N + 48] = acc3[v];
    }
}

// ---------------------------------------------------------------------------
// Elementwise / permute / reduction helpers
// ---------------------------------------------------------------------------
__global__ void k_gather(const float* __restrict__ p, const int* __restrict__ idx,
                         float* __restrict__ A) {
    int i = blockIdx.x * blockDim.x + threadIdx.x;
    if (i < 512) A[i] = p[idx[i]];
}

// a[e,a,f,b,g,c,h,d] = sum_m A[m,e,f,g,h]*A[m,a,b,c,d]   (un-normalized)
__global__ void k_dlayer(const float* __restrict__ A, float* __restrict__ a) {
    int o = blockIdx.x * blockDim.x + threadIdx.x;
    if (o >= 65536) return;
    int d = o & 3, h = (o >> 2) & 3, c = (o >> 4) & 3, g = (o >> 6) & 3;
    int b = (o >> 8) & 3, f = (o >> 10) & 3, aa = (o >> 12) & 3, e = (o >> 14) & 3;
    float s = 0.f;
    for (int m = 0; m < 2; ++m)
        s += A[(((m * 4 + e) * 4 + f) * 4 + g) * 4 + h] *
             A[(((m * 4 + aa) * 4 + b) * 4 + c) * 4 + d];
    a[o] = s;
}

__global__ void k_sumsq(const float* __restrict__ x, int n, float* __restrict__ out) {
    __shared__ float sm[256];
    float s = 0.f;
    for (int i = threadIdx.x; i < n; i += 256) { float v = x[i]; s += v * v; }
    sm[threadIdx.x] = s; __syncthreads();
    for (int off = 128; off > 0; off >>= 1) {
        if ((int)threadIdx.x < off) sm[threadIdx.x] += sm[threadIdx.x + off];
        __syncthreads();
    }
    if (threadIdx.x == 0) out[0] = sm[0];
}

__global__ void k_scale(float* __restrict__ x, int n, const float* __restrict__ ss) {
    int i = blockIdx.x * blockDim.x + threadIdx.x;
    if (i >= n) return;
    x[i] *= rsqrtf(ss[0] + 1e-30f);
}

// dst = src + src^T  (square n x n); the 0.5 factor is absorbed by normalization
__global__ void k_addT(const float* __restrict__ src, float* __restrict__ dst, int n) {
    int i = blockIdx.x * blockDim.x + threadIdx.x;
    if (i >= n * n) return;
    int r = i / n, c = i % n;
    dst[i] = src[i] + src[c * n + r];
}

// T(64,16,64): dst[x,u,y] = src[x,u,y] + src[y,u,x]
__global__ void k_symT3(const float* __restrict__ src, float* __restrict__ dst) {
    int i = blockIdx.x * blockDim.x + threadIdx.x;
    if (i >= 65536) return;
    int y = i & 63, u = (i >> 6) & 15, x = i >> 10;
    dst[i] = src[i] + src[(y * 16 + u) * 64 + x];
}

__global__ void k_transpose(const float* __restrict__ src, float* __restrict__ dst,
                            int M, int N) {
    int i = blockIdx.x * blockDim.x + threadIdx.x;
    if (i >= M * N) return;
    int r = i / N, c = i % N;
    dst[(size_t)c * M + r] = src[i];
}

__global__ void k_copy(const float* __restrict__ s, float* __restrict__ d, int n) {
    int i = blockIdx.x * blockDim.x + threadIdx.x;
    if (i < n) d[i] = s[i];
}

// R2[(f,g),(h,i)](1024x1024) -> R2P[(g,i),(f,h)](4096x256)
__global__ void k_perm_R2(const float* __restrict__ src, float* __restrict__ dst) {
    int j = blockIdx.x * blockDim.x + threadIdx.x;
    if (j >= 1048576) return;
    int row = j >> 8, col = j & 255;
    int g = row >> 6, i2 = row & 63, f = col >> 4, h = col & 15;
    dst[j] = src[(f * 64 + g) * 1024 + h * 64 + i2];
}

// Out4[(g,i),(c,d)](4096x256) -> Rho[(g,d),(i,c)](1024x1024)
__global__ void k_perm_Out4(const float* __restrict__ src, float* __restrict__ dst) {
    int j = blockIdx.x * blockDim.x + threadIdx.x;
    if (j >= 1048576) return;
    int row = j >> 10, col = j & 1023;
    int g = row >> 4, d = row & 15, i2 = col >> 4, c = col & 15;
    dst[j] = src[(g * 64 + i2) * 256 + c * 16 + d];
}

// P(1024x64) -> Px[x,(c,j)](64x1024) with P3[x,c,j]=P[x*16+c, j]
__global__ void k_P_to_Px(const float* __restrict__ P, float* __restrict__ Px) {
    int i = blockIdx.x * blockDim.x + threadIdx.x;
    if (i >= 65536) return;
    int x = i >> 10, c = (i >> 6) & 15, j = i & 63;
    Px[i] = P[(x * 16 + c) * 64 + j];
}

// Tn1[(h,i),(c,j)](1024x1024) -> A4[(i,j),(h,c)](4096x256)
__global__ void k_perm_Tn1(const float* __restrict__ src, float* __restrict__ dst) {
    int j = blockIdx.x * blockDim.x + threadIdx.x;
    if (j >= 1048576) return;
    int row = j >> 8, col = j & 255;
    int i2 = row >> 6, jj = row & 63, h = col >> 4, c = col & 15;
    dst[j] = src[(h * 64 + i2) * 1024 + c * 64 + jj];
}

// Tn2[(i,j),(r,s)](4096x256) -> A5[(j,s),(i,r)](1024x1024)
__global__ void k_perm_Tn2(const float* __restrict__ src, float* __restrict__ dst) {
    int j = blockIdx.x * blockDim.x + threadIdx.x;
    if (j >= 1048576) return;
    int row = j >> 10, col = j & 1023;
    int jj = row >> 4, s = row & 15, i2 = col >> 4, r = col & 15;
    dst[j] = src[(i2 * 64 + jj) * 256 + r * 16 + s];
}

// amat2[(h,c),(r,s)] = a[c,h,r,s]
__global__ void k_build_amat2(const float* __restrict__ a, float* __restrict__ b) {
    int j = blockIdx.x * blockDim.x + threadIdx.x;
    if (j >= 65536) return;
    int row = j >> 8, col = j & 255;
    int h = row >> 4, c = row & 15, r = col >> 4, s = col & 15;
    b[j] = a[((c * 16 + h) * 16 + r) * 16 + s];
}

// TdB[(u,v),(g,h,m,n)](256x1024) = Td[v,u,g,h,m,n] from A(2,4,4,4,4)
__global__ void k_build_TdB(const float* __restrict__ A, float* __restrict__ B) {
    int j = blockIdx.x * blockDim.x + threadIdx.x;
    if (j >= 262144) return;
    int row = j >> 10, col = j & 1023;
    int u = row >> 4, v = row & 15;
    int g2 = col >> 6, h2 = (col >> 2) & 15, m = (col >> 1) & 1, n = col & 1;
    int e = v >> 2, a2 = v & 3, f = u >> 2, b2 = u & 3;
    int g1 = g2 >> 2, c1 = g2 & 3, h1 = h2 >> 2, d1 = h2 & 3;
    B[j] = A[(((m * 4 + e) * 4 + f) * 4 + g1) * 4 + h1] *
           A[(((n * 4 + a2) * 4 + b2) * 4 + c1) * 4 + d1];
}

// EL1[(x,u),(v,w)](1024x1024) -> A2[(x,w),(u,v)](4096x256)
__global__ void k_perm_EL1(const float* __restrict__ src, float* __restrict__ dst) {
    int j = blockIdx.x * blockDim.x + threadIdx.x;
    if (j >= 1048576) return;
    int row = j >> 8, col = j & 255;
    int x = row >> 6, w = row & 63, u = col >> 4, v = col & 15;
    dst[j] = src[(x * 16 + u) * 1024 + v * 64 + w];
}

// EL2[(x,w),(g,h,m,n)](4096x1024) -> A3[(w,h,m,n),(x,g)](4096x1024)
__global__ void k_perm_EL2(const float* __restrict__ src, float* __restrict__ dst) {
    int j = blockIdx.x * blockDim.x + threadIdx.x;
    if (j >= 4194304) return;
    int row = j >> 10, col = j & 1023;
    int w = row >> 6, h = (row >> 2) & 15, mn = row & 3;
    int x = col >> 4, g = col & 15;
    dst[j] = src[(size_t)(x * 64 + w) * 1024 + g * 64 + h * 4 + mn];
}

// Deterministic pseudo-random init for subspace iteration
__global__ void k_initQ(float* __restrict__ Q) {
    int i = blockIdx.x * blockDim.x + threadIdx.x;
    if (i >= 65536) return;
    unsigned u = (unsigned)i * 2654435761u ^ 0x9e3779b9u;
    u ^= u >> 16; u *= 0x85ebca6bu; u ^= u >> 13;
    Q[i] = ((float)(u & 0xFFFF) / 32768.0f) - 1.0f;
}

// Cholesky of G (64x64 SPD) then B[j*64+i] = Linv[i][j]  (so Q = Z2 @ B == Z2 @ Linv^T)
__global__ void k_cholinv(const float* __restrict__ G, float* __restrict__ B) {
    if (threadIdx.x != 0 || blockIdx.x != 0) return;
    float L[64][64];
    float tr = 0.f;
    for (int i = 0; i < 64; ++i) tr += G[i * 64 + i];
    float eps = 1e-10f * tr + 1e-30f;
    for (int i = 0; i < 64; ++i)
        for (int j2 = 0; j2 <= i; ++j2) L[i][j2] = G[i * 64 + j2];
    for (int k = 0; k < 64; ++k) {
        float d = L[k][k] + eps;
        for (int j2 = 0; j2 < k; ++j2) d -= L[k][j2] * L[k][j2];
        d = sqrtf(fmaxf(d, 1e-30f));
        L[k][k] = d;
        float inv = 1.f / d;
        for (int i = k + 1; i < 64; ++i) {
            float s = L[i][k];
            for (int j2 = 0; j2 < k; ++j2) s -= L[i][j2] * L[k][j2];
            L[i][k] = s * inv;
        }
    }
    float X[64][64];
    for (int c = 0; c < 64; ++c)
        for (int i = 0; i < 64; ++i) {
            if (i < c) { X[i][c] = 0.f; continue; }
            float s = (i == c) ? 1.f : 0.f;
            for (int j2 = c; j2 < i; ++j2) s -= L[i][j2] * X[j2][c];
            X[i][c] = s / L[i][i];
        }
    for (int j2 = 0; j2 < 64; ++j2)
        for (int i = 0; i < 64; ++i) B[j2 * 64 + i] = X[i][j2];
}

// 64x64 symmetric Jacobi eigensolver in LDS; V columns sorted by |lambda| desc.
__global__ __launch_bounds__(64) void k_jacobi(const float* __restrict__ Sin,
                                               float* __restrict__ Vout) {
    __shared__ float S[64][64];
    __shared__ float V[64][64];
    __shared__ float cs[2];
    __shared__ int order[64];
    int t = threadIdx.x;
    for (int j = 0; j < 64; ++j) S[j][t] = Sin[j * 64 + t];
    for (int j = 0; j < 64; ++j) V[j][t] = (j == t) ? 1.f : 0.f;
    __syncthreads();
    for (int sweep = 0; sweep < 8; ++sweep) {
        for (int p = 0; p < 63; ++p)
            for (int q = p + 1; q < 64; ++q) {
                if (t == 0) {
                    float apq = S[p][q], c = 1.f, s = 0.f;
                    if (fabsf(apq) > 1e-12f) {
                        float tau = (S[q][q] - S[p][p]) / (2.f * apq);
                        float tt = (tau >= 0.f ? 1.f : -1.f) /
                                   (fabsf(tau) + sqrtf(1.f + tau * tau));
                        c = rsqrtf(1.f + tt * tt); s = tt * c;
                    }
                    cs[0] = c; cs[1] = s;
                }
                __syncthreads();
                float c = cs[0], s = cs[1];
                float Sp = S[p][t], Sq = S[q][t];
                S[p][t] = c * Sp - s * Sq;
                S[q][t] = s * Sp + c * Sq;
                __syncthreads();
                Sp = S[t][p]; Sq = S[t][q];
                S[t][p] = c * Sp - s * Sq;
                S[t][q] = s * Sp + c * Sq;
                float Vp = V[t][p], Vq = V[t][q];
                V[t][p] = c * Vp - s * Vq;
                V[t][q] = s * Vp + c * Vq;
                __syncthreads();
            }
    }
    if (t == 0) {
        float ev[64];
        for (int i = 0; i < 64; ++i) { ev[i] = fabsf(S[i][i]); order[i] = i; }
        for (int i = 0; i < 63; ++i) {
            int best = i;
            for (int j = i + 1; j < 64; ++j)
                if (ev[order[j]] > ev[order[best]]) best = j;
            int tmp = order[i]; order[i] = order[best]; order[best] = tmp;
        }
    }
    __syncthreads();
    for (int j = 0; j < 64; ++j) Vout[t * 64 + j] = V[t][order[j]];
}

// Rho16[mn,m'n'] = sum_{w,h,z} EL3[(w,h,mn),z] * EL3[(w,h,m'n'),z]
__global__ __launch_bounds__(256) void k_rho16(const float* __restrict__ EL3,
                                               float* __restrict__ R16) {
    __shared__ float sm[16 * 256];
    float acc[16];
#pragma unroll
    for (int i = 0; i < 16; ++i) acc[i] = 0.f;
    for (int wh = threadIdx.x; wh < 1024; wh += 256) {
        const float* base = EL3 + (size_t)wh * 4 * 64;
        for (int z = 0; z < 64; ++z) {
            float e0 = base[z], e1 = base[64 + z], e2 = base[128 + z], e3 = base[192 + z];
            float e[4] = {e0, e1, e2, e3};
#pragma unroll
            for (int i2 = 0; i2 < 4; ++i2)
#pragma unroll
                for (int j2 = 0; j2 < 4; ++j2) acc[i2 * 4 + j2] += e[i2] * e[j2];
        }
    }
    for (int i = 0; i < 16; ++i) sm[i * 256 + threadIdx.x] = acc[i];
    __syncthreads();
    for (int off = 128; off > 0; off >>= 1) {
        if ((int)threadIdx.x < off)
            for (int i = 0; i < 16; ++i) sm[i * 256 + threadIdx.x] += sm[i * 256 + threadIdx.x + off];
        __syncthreads();
    }
    if (threadIdx.x < 16) R16[threadIdx.x] = sm[threadIdx.x * 256];
}

__global__ void k_loss(const float* __restrict__ R16, const float* __restrict__ H,
                       float* __restrict__ out) {
    if (threadIdx.x || blockIdx.x) return;
    float num = 0.f, den = 0.f;
    for (int i = 0; i < 4; ++i)
        for (int j = 0; j < 4; ++j) {
            float rs = 0.5f * (R16[i * 4 + j] + R16[j * 4 + i]);
            num += rs * H[j * 4 + i];
            if (i == j) den += rs;
        }
    out[0] = num / den;
}

// ---------------------------------------------------------------------------
// Host orchestration
// ---------------------------------------------------------------------------
extern "C" void kernel_launch(void* const* d_in, const int* in_sizes, int n_in,
                              void* d_out, int out_size, void* d_ws, size_t ws_size,
                              hipStream_t stream) {
    const float* Hm   = (const float*)d_in[0];
    const float* prm  = (const float*)d_in[1];
    const int*   midx = (const int*)d_in[2];
    const float* Cin  = (const float*)d_in[3];
    const float* Tin  = (const float*)d_in[4];
    float* out = (float*)d_out;

    float* w = (float*)d_ws;
    size_t off = 0;
    auto alloc = [&](size_t n) { float* p = w + off; off += (n + 63) & ~(size_t)63; return p; };

    float* A     = alloc(512);
    float* aT    = alloc(65536);     // doubles as amat (256x256)
    float* amat2 = alloc(65536);
    float* Cb0   = alloc(4096);
    float* Cb1   = alloc(4096);
    float* Tb0   = alloc(65536);
    float* Tb1   = alloc(65536);
    float* R1    = alloc(65536);     // doubles as CE in energy phase
    float* R1T   = alloc(65536);
    float* R2    = alloc(1048576);   // doubles as EL1
    float* R2P   = alloc(1048576);   // doubles as A2
    float* OUT4  = alloc(1048576);
    float* RHOR  = alloc(1048576);
    float* RHOS  = alloc(1048576);
    float* Q     = alloc(65536);
    float* Z     = alloc(65536);
    float* Z2    = alloc(65536);
    float* TRS   = alloc(65536);     // transpose scratch (Z2T / QT / PT / TT)
    float* G     = alloc(4096);
    float* LINV  = alloc(4096);
    float* Sm    = alloc(4096);
    float* Vm    = alloc(4096);
    float* Wb    = alloc(65536);
    float* P     = alloc(65536);
    float* PX    = alloc(65536);
    float* TN1   = alloc(1048576);
    float* A4    = alloc(1048576);
    float* TN2   = alloc(1048576);
    float* A5    = alloc(1048576);
    float* TNU   = alloc(65536);
    float* W2    = alloc(65536);
    float* CNU   = alloc(4096);
    float* scal  = alloc(64);
    float* TDB   = alloc(262144);
    float* EL2   = alloc(4194304);
    float* A3    = alloc(4194304);
    float* EL3   = alloc(262144);
    float* R16   = alloc(64);

    auto gemm = [&](const float* Am, const float* Bm, float* Cm, int M, int N, int K) {
        int blocks = (M >> 6) * (N >> 6);
        k_gemm<<<dim3(blocks), dim3(128), 0, stream>>>(Am, Bm, Cm, M, N, K);
    };
    auto normalize = [&](float* buf, int n) {
        k_sumsq<<<dim3(1), dim3(256), 0, stream>>>(buf, n, scal);
        k_scale<<<dim3((n + 255) / 256), dim3(256), 0, stream>>>(buf, n, scal);
    };
    auto g256 = [](int n) { return dim3((n + 255) / 256); };

    // ---- setup: A, double layer a, amat2, C0, T0 ----
    k_gather<<<dim3(2), dim3(256), 0, stream>>>(prm, midx, A);
    k_dlayer<<<g256(65536), dim3(256), 0, stream>>>(A, aT);
    normalize(aT, 65536);
    k_build_amat2<<<g256(65536), dim3(256), 0, stream>>>(aT, amat2);
    k_addT<<<g256(4096), dim3(256), 0, stream>>>(Cin, Cb0, 64);
    normalize(Cb0, 4096);
    k_symT3<<<g256(65536), dim3(256), 0, stream>>>(Tin, Tb0);
    normalize(Tb0, 65536);

    float *C = Cb0, *Cn = Cb1, *T = Tb0, *Tn = Tb1;

    // ---- CTM loop ----
    for (int it = 0; it < 15; ++it) {
        // Rho construction
        gemm(C, T, R1, 64, 1024, 64);                                  // R1[e,(f,g)]
        k_transpose<<<g256(65536), dim3(256), 0, stream>>>(R1, R1T, 64, 1024);
        gemm(R1T, T, R2, 1024, 1024, 64);                              // R2[(f,g),(h,i)]
        k_perm_R2<<<g256(1048576), dim3(256), 0, stream>>>(R2, R2P);
        gemm(R2P, aT, OUT4, 4096, 256, 256);                           // Out4[(g,i),(c,d)]
        k_perm_Out4<<<g256(1048576), dim3(256), 0, stream>>>(OUT4, RHOR);
        k_addT<<<g256(1048576), dim3(256), 0, stream>>>(RHOR, RHOS, 1024);
        normalize(RHOS, 1048576);

        // top-64 singular subspace of symmetric RHOS via subspace iteration
        k_initQ<<<g256(65536), dim3(256), 0, stream>>>(Q);
        for (int s = 0; s < 12; ++s) {
            gemm(RHOS, Q, Z, 1024, 64, 1024);
            gemm(RHOS, Z, Z2, 1024, 64, 1024);
            k_transpose<<<g256(65536), dim3(256), 0, stream>>>(Z2, TRS, 1024, 64);
            gemm(TRS, Z2, G, 64, 64, 1024);                            // G = Z2^T Z2
            k_cholinv<<<dim3(1), dim3(1), 0, stream>>>(G, LINV);
            gemm(Z2, LINV, Q, 1024, 64, 64);                           // Q = Z2 Linv^T
        }
        // Rayleigh-Ritz + sort by |lambda|
        gemm(RHOS, Q, Wb, 1024, 64, 1024);
        k_transpose<<<g256(65536), dim3(256), 0, stream>>>(Q, TRS, 1024, 64);
        gemm(TRS, Wb, Sm, 64, 64, 1024);                               // S = Q^T Rho Q
        k_jacobi<<<dim3(1), dim3(64), 0, stream>>>(Sm, Vm);
        gemm(Q, Vm, P, 1024, 64, 64);                                  // P

        // Cn = P^T Rho P (sym + norm)
        gemm(RHOS, P, W2, 1024, 64, 1024);
        k_transpose<<<g256(65536), dim3(256), 0, stream>>>(P, TRS, 1024, 64);
        gemm(TRS, W2, CNU, 64, 64, 1024);
        k_addT<<<g256(4096), dim3(256), 0, stream>>>(CNU, Cn, 64);
        normalize(Cn, 4096);

        // Tn
        k_transpose<<<g256(65536), dim3(256), 0, stream>>>(T, TRS, 64, 1024);
        k_P_to_Px<<<g256(65536), dim3(256), 0, stream>>>(P, PX);
        gemm(TRS, PX, TN1, 1024, 1024, 64);                            // Tn1[(h,i),(c,j)]
        k_perm_Tn1<<<g256(1048576), dim3(256), 0, stream>>>(TN1, A4);
        gemm(A4, amat2, TN2, 4096, 256, 256);                          // Tn2[(i,j),(r,s)]
        k_perm_Tn2<<<g256(1048576), dim3(256), 0, stream>>>(TN2, A5);
        gemm(A5, P, TNU, 1024, 64, 1024);                              // Tn[j,s,k]
        k_symT3<<<g256(65536), dim3(256), 0, stream>>>(TNU, Tn);
        normalize(Tn, 65536);

        float* tmp;
        tmp = C; C = Cn; Cn = tmp;
        tmp = T; T = Tn; Tn = tmp;
    }

    // ---- energy with final C, T ----
    gemm(C, T, R1, 64, 1024, 64);                                      // CE[y,(v,w)]
    gemm(T, R1, R2, 1024, 1024, 64);                                   // EL1[(x,u),(v,w)]
    k_perm_EL1<<<g256(1048576), dim3(256), 0, stream>>>(R2, R2P);      // A2[(x,w),(u,v)]
    k_build_TdB<<<g256(262144), dim3(256), 0, stream>>>(A, TDB);
    gemm(R2P, TDB, EL2, 4096, 1024, 256);                              // EL2[(x,w),(g,h,m,n)]
    k_perm_EL2<<<dim3((4194304 + 255) / 256), dim3(256), 0, stream>>>(EL2, A3);
    gemm(A3, R1, EL3, 4096, 64, 1024);                                 // EL3[(w,h,m,n),z], CE as (1024x64)
    k_rho16<<<dim3(1), dim3(256), 0, stream>>>(EL3, R16);
    k_loss<<<dim3(1), dim3(1), 0, stream>>>(R16, Hm, out);

    // ---- outputs: (loss, Cf, Tf) ----
    k_copy<<<g256(4096), dim3(256), 0, stream>>>(C, out + 1, 4096);
    k_copy<<<g256(65536), dim3(256), 0, stream>>>(T, out + 1 + 4096, 65536);
}